// FractalFieldCalculator_773094113764
// MI455X (gfx1250) — compile-verified
//
#include <hip/hip_runtime.h>
#include <stdint.h>

#define BATCH     4096
#define EMBED     4096
#define MAX_TERMS 20
#define TPB       256
#define WS_FACTOR 32   // float offset of factor[] inside ws (coeff[] at 0)

typedef float v4f __attribute__((ext_vector_type(4)));

__device__ __forceinline__ uint32_t pcg_hash(uint32_t v) {
    v = v * 747796405u + 2891336453u;
    uint32_t w = ((v >> ((v >> 28u) + 4u)) ^ v) * 277803737u;
    return (w >> 22u) ^ w;
}

// single counter-based normal (only used on rare NaN/bump paths)
__device__ __forceinline__ float hash_normal(uint32_t seed, uint32_t idx) {
    uint32_t a = pcg_hash(seed ^ (idx * 0x9E3779B9u));
    uint32_t b = pcg_hash(a + 0x85EBCA6Bu);
    float u1 = (float)((a >> 8) + 1u) * (1.0f / 16777216.0f);
    float u2 = (float)(b >> 8) * (1.0f / 16777216.0f);
    return sqrtf(-2.0f * __logf(u1)) * __cosf(6.28318530718f * u2);
}

// paired Box-Muller: two normals from one (log,sqrt,sincos) + 2 hashes
__device__ __forceinline__ void hash_normal2(uint32_t seed, uint32_t idx,
                                             float* n0, float* n1) {
    uint32_t a = pcg_hash(seed ^ (idx * 0x9E3779B9u));
    uint32_t b = pcg_hash(a + 0x85EBCA6Bu);
    float u1 = (float)((a >> 8) + 1u) * (1.0f / 16777216.0f);
    float u2 = (float)(b >> 8) * (1.0f / 16777216.0f);
    float r  = sqrtf(-2.0f * __logf(u1));
    float sn, cs;
    __sincosf(6.28318530718f * u2, &sn, &cs);
    *n0 = r * cs;
    *n1 = r * sn;
}

// --------- setup: per-dispatch coeff[20] and per-column factor[4096] ---------
__global__ void fractal_setup_kernel(const float* __restrict__ lam,
                                     const int*   __restrict__ time_p,
                                     float*       __restrict__ ws) {
    const int i = blockIdx.x * blockDim.x + threadIdx.x;
    const int t = *time_p;
    if (i < MAX_TERMS) {
        // coeff[k] = lam[k] * cos(2*pi*ln(k+1)) / k!
        float fact = 1.0f;
        for (int m = 2; m <= i; ++m) fact *= (float)m;
        float ct = (i == 0) ? 1.0f
                            : __cosf(6.283185307179586f * __logf((float)(i + 1)));
        ws[i] = lam[i] * ct / fact;
    }
    if (i < EMBED) {
        float temporal = __powf((float)t + 1.0f, 0.2f);        // (t+1)^(H-0.5)
        float freq     = ((float)i + 1.0f) * (1.0f / (float)EMBED);
        float spatial  = __sinf(6.28318530718f * freq * (float)t);
        ws[WS_FACTOR + i] = temporal * (1.0f + 0.3f * spatial) * 0.1f;
    }
}

// --------------------------------- main ------------------------------------
__global__ __launch_bounds__(TPB)
void fractal_field_kernel(const float* __restrict__ psi,
                          const float* __restrict__ ws,
                          const int*   __restrict__ time_p,
                          float*       __restrict__ out)
{
    __shared__ __align__(16) float s_field[EMBED];   // 16 KB row tile
    __shared__ float s_red[TPB / 32];
    __shared__ float s_bcast;

    const int row  = blockIdx.x;
    const int tid  = threadIdx.x;
    const int lane = tid & 31;
    const int wid  = tid >> 5;
    const float* rowp   = psi + (size_t)row * EMBED;
    float*       outp   = out + (size_t)row * EMBED;
    const float* factor = ws + WS_FACTOR;

    // ---- Phase A: async DMA psi row into LDS (CDNA5 async global->LDS) ----
    const uint32_t lds_base = (uint32_t)(uintptr_t)(&s_field[0]);
#pragma unroll
    for (int q = 0; q < 4; ++q) {
        uint32_t lds_off = lds_base + (uint32_t)((q * TPB + tid) * 16);
        const float* gp  = rowp + (q * TPB + tid) * 4;
        asm volatile("global_load_async_to_lds_b128 %0, %1, off"
                     :: "v"(lds_off), "v"(gp) : "memory");
    }

    // Overlap DMA latency: uniform scalar loads of polynomial coefficients
    float C[MAX_TERMS];
#pragma unroll
    for (int k = 0; k < MAX_TERMS; ++k) C[k] = ws[k];   // s_load (uniform)
    const uint32_t nseed = (uint32_t)(*time_p * 1000);

    asm volatile("s_wait_asynccnt 0x0" ::: "memory");   // my DMA landed
    __syncthreads();                                    // everyone's visible

    // ---- Phase B: field = -gradV(psi) + noise ----
    float ss = 0.0f;
#pragma unroll
    for (int q = 0; q < 4; ++q) {
        const int c = (q * TPB + tid) * 4;
        v4f p4 = *reinterpret_cast<const v4f*>(&s_field[c]);
        v4f f4 = *reinterpret_cast<const v4f*>(factor + c);          // L2-hot
        float p[4]   = {p4.x, p4.y, p4.z, p4.w};
        float fac[4] = {f4.x, f4.y, f4.z, f4.w};
        float n[4];
        hash_normal2(nseed, (uint32_t)(row * EMBED + c),     &n[0], &n[1]);
        hash_normal2(nseed, (uint32_t)(row * EMBED + c + 2), &n[2], &n[3]);
        float g[4];
#pragma unroll
        for (int e = 0; e < 4; ++e) {
            float base = p[e] + 1e-8f;
            float acc  = C[MAX_TERMS - 1];
#pragma unroll
            for (int k = MAX_TERMS - 2; k >= 0; --k) acc = fmaf(acc, base, C[k]);
            float v = fmaf(n[e], fac[e], -acc);
            g[e] = v;
            ss   = fmaf(v, v, ss);
        }
        v4f o4 = {g[0], g[1], g[2], g[3]};
        *reinterpret_cast<v4f*>(&s_field[c]) = o4;
    }

    // ---- Phase C: row norm -> clip scale (folded into linear smoothing) ----
#pragma unroll
    for (int off = 16; off > 0; off >>= 1) ss += __shfl_down(ss, off, 32);
    if (lane == 0) s_red[wid] = ss;
    __syncthreads();
    if (tid == 0) {
        float tot = 0.0f;
        for (int w = 0; w < TPB / 32; ++w) tot += s_red[w];
        float mag = sqrtf(tot);
        s_bcast = (mag > 10.0f) ? (10.0f / (mag + 1e-8f)) : 1.0f;
    }
    __syncthreads();
    const float scale = s_bcast;

    // ---- Phase D: circular 3-tap smooth + NaN/Inf replace + second norm ----
    float ss2 = 0.0f;
    float ov[16];
#pragma unroll
    for (int q = 0; q < 4; ++q) {
        const int c = (q * TPB + tid) * 4;
        v4f m4      = *reinterpret_cast<const v4f*>(&s_field[c]);
        float left  = s_field[(c + EMBED - 1) & (EMBED - 1)];
        float right = s_field[(c + 4) & (EMBED - 1)];
        float m[6]  = {left, m4.x, m4.y, m4.z, m4.w, right};
#pragma unroll
        for (int e = 0; e < 4; ++e) {
            float v = scale * (0.25f * m[e] + 0.5f * m[e + 1] + 0.25f * m[e + 2]);
            uint32_t bits = __float_as_uint(v);
            if ((bits & 0x7F800000u) == 0x7F800000u)       // NaN or Inf (rare)
                v = hash_normal(123u, (uint32_t)(row * EMBED + c + e)) * 0.01f;
            ov[q * 4 + e] = v;
            ss2 = fmaf(v, v, ss2);
        }
    }

    // ---- Phase E: second row norm -> conditional bump ----
#pragma unroll
    for (int off = 16; off > 0; off >>= 1) ss2 += __shfl_down(ss2, off, 32);
    __syncthreads();                    // protect s_red / s_bcast reuse
    if (lane == 0) s_red[wid] = ss2;
    __syncthreads();
    if (tid == 0) {
        float tot = 0.0f;
        for (int w = 0; w < TPB / 32; ++w) tot += s_red[w];
        s_bcast = sqrtf(tot);
    }
    __syncthreads();
    const bool bump = (s_bcast < 1e-6f);

    // ---- Phase F: streaming 128-bit non-temporal stores ----
#pragma unroll
    for (int q = 0; q < 4; ++q) {
        const int c = (q * TPB + tid) * 4;
        v4f o = {ov[q * 4 + 0], ov[q * 4 + 1], ov[q * 4 + 2], ov[q * 4 + 3]};
        if (bump) {                                        // uniform, rare
            o.x += hash_normal(456u, (uint32_t)(row * EMBED + c + 0)) * 1e-6f;
            o.y += hash_normal(456u, (uint32_t)(row * EMBED + c + 1)) * 1e-6f;
            o.z += hash_normal(456u, (uint32_t)(row * EMBED + c + 2)) * 1e-6f;
            o.w += hash_normal(456u, (uint32_t)(row * EMBED + c + 3)) * 1e-6f;
        }
        __builtin_nontemporal_store(o, reinterpret_cast<v4f*>(outp + c));
    }
}

extern "C" void kernel_launch(void* const* d_in, const int* in_sizes, int n_in,
                              void* d_out, int out_size, void* d_ws, size_t ws_size,
                              hipStream_t stream) {
    const float* psi    = (const float*)d_in[0];   // [4096,4096] f32
    const float* lam    = (const float*)d_in[1];   // [20] f32
    const int*   time_p = (const int*)d_in[2];     // scalar int
    float*       out    = (float*)d_out;           // [4096,4096] f32
    float*       ws     = (float*)d_ws;            // >= (32+4096)*4 bytes

    hipLaunchKernelGGL(fractal_setup_kernel, dim3(EMBED / TPB), dim3(TPB), 0, stream,
                       lam, time_p, ws);
    hipLaunchKernelGGL(fractal_field_kernel, dim3(BATCH), dim3(TPB), 0, stream,
                       psi, (const float*)ws, time_p, out);
}